// DetectionLoss_3839700762852
// MI455X (gfx1250) — compile-verified
//
#include <hip/hip_runtime.h>
#include <cstdint>

// ---------------------------------------------------------------------------
// Detection loss (SSD-style matching + DIoU + focal w/ hard-negative mining)
// for MI455X (gfx1250). VALU-bound in the 16x65536x64 pairwise IoU; the
// reference's per-image full sort is replaced by a 4096-bucket float-bits
// histogram select (exact above the boundary bucket, mean-corrected within).
// CDNA5 paths: TDM tensor_load_to_lds stages GT tiles into LDS (match + loss
// kernels), v_wmma_f32_16x16x4_f32 performs the wave-level sum reductions,
// ds_max_u64 LDS atomics do the per-GT argmax. Hot-loop divisions use
// v_rcp_f32 (IoU already carries +1e-7 slop), focal uses v_exp/v_log.
// ---------------------------------------------------------------------------

typedef __attribute__((ext_vector_type(2))) float        v2f;
typedef __attribute__((ext_vector_type(8))) float        v8f;
typedef __attribute__((ext_vector_type(4))) unsigned int v4u;
typedef __attribute__((ext_vector_type(8))) int          v8i;
typedef __attribute__((ext_vector_type(4))) int          v4i;

#define NUM_B   16
#define NUM_A   65536
#define NUM_G   64
#define NBUCKET 4096
#define EPSF    1e-7f

// ---- workspace layout (bytes) ----
#define PERGT_OFF   0u                              // u64 [B][G]      (8192)
#define NUMPOS_OFF  8192u                           // f32 [B]
#define LOCSUM_OFF  8256u                           // f32 [B]
#define POSFOC_OFF  8320u                           // f32 [B]
#define CONF_OFF    8384u                           // f32 [B]
#define HISTC_OFF   16384u                          // u32 [B][4096]   (262144)
#define HISTS_OFF   278528u                         // f32 [B][4096]   (262144)
#define PFORCE_OFF  540672u                         // u8  [B][A]      (1048576)
#define BESTP_OFF   1589248u                        // u64 [B][A]      (8388608)
#define ZERO_WORDS  (1589248u / 4u)                 // zero everything before BESTP

#if defined(__has_builtin)
# if __has_builtin(__builtin_amdgcn_tensor_load_to_lds) && __has_builtin(__builtin_amdgcn_s_wait_tensorcnt)
#  define HAVE_TDM 1
# endif
# if __has_builtin(__builtin_amdgcn_wmma_f32_16x16x4_f32)
#  define HAVE_WMMA_F32 1
# endif
# if __has_builtin(__builtin_amdgcn_rcpf)
#  define HAVE_RCP 1
# endif
#endif

__device__ __forceinline__ float fast_rcp(float x) {
#if defined(HAVE_RCP)
  return __builtin_amdgcn_rcpf(x);        // v_rcp_f32, ~1 ULP
#else
  return 1.0f / x;
#endif
}

// Wave32 sum reduction via V_WMMA_F32_16X16X4_F32:
//   A[m][0] = p_m (lanes 0-15), A[m][2] = p_{m+16} (lanes 16-31), K=1,3 zero.
//   B = ones(4x16)  =>  D[m][n] = p_m + p_{m+16}.
//   Sum D's 8 VGPRs per lane, then add across the lane^16 halves.
__device__ __forceinline__ float wave_sum32(float p) {
#if defined(HAVE_WMMA_F32)
  v2f a;  a.x = p;    a.y = 0.0f;
  v2f bo; bo.x = 1.0f; bo.y = 1.0f;
  v8f c = {0.f, 0.f, 0.f, 0.f, 0.f, 0.f, 0.f, 0.f};
  c = __builtin_amdgcn_wmma_f32_16x16x4_f32(false, a, false, bo, (short)0, c,
                                            false, false);
  float s = ((c[0] + c[1]) + (c[2] + c[3])) + ((c[4] + c[5]) + (c[6] + c[7]));
  s += __shfl_xor(s, 16);
  return s;
#else
  for (int off = 16; off > 0; off >>= 1) p += __shfl_xor(p, off);
  return p;
#endif
}

// Stage the 256-float GT tile for image `b` into LDS with the Tensor Data
// Mover (wave 0 issues the descriptor; caller must __syncthreads()).
__device__ __forceinline__ void stage_gt_tile(const float* __restrict__ gptr,
                                              float* sGT, int tid) {
  if (tid < 32) {
#if defined(HAVE_TDM)
    unsigned long long gaddr = (unsigned long long)(uintptr_t)gptr;
    unsigned ldsOff = (unsigned)(uintptr_t)(&sGT[0]);
    v4u g0;
    g0.x = 1u;                                   // count=1, no gather
    g0.y = ldsOff;                               // lds_addr
    g0.z = (unsigned)gaddr;                      // global_addr[95:64]
    g0.w = ((unsigned)(gaddr >> 32) & 0x01FFFFFFu) | (2u << 30); // hi | type=2
    v8i g1;
    g1[0] = (int)(2u << 16);     // wg_mask=0, data_size=4B
    g1[1] = (int)(256u << 16);   // tensor_dim0 = 256 (low16 in [31:16])
    g1[2] = (int)(1u << 16);     // tensor_dim0 hi=0; tensor_dim1 = 1
    g1[3] = (int)(256u << 16);   // tensor_dim1 hi=0; tile_dim0 = 256
    g1[4] = 1;                   // tile_dim1 = 1; tile_dim2 = 0
    g1[5] = 256;                 // tensor_dim0_stride low32
    g1[6] = (int)(256u << 16);   // stride0 hi=0; tensor_dim1_stride low16
    g1[7] = 0;
    v4i gz = {0, 0, 0, 0};
# if defined(__clang_major__) && (__clang_major__ >= 23)
    v8i g4z = {0, 0, 0, 0, 0, 0, 0, 0};
    __builtin_amdgcn_tensor_load_to_lds(g0, g1, gz, gz, g4z, 0);
# else
    __builtin_amdgcn_tensor_load_to_lds(g0, g1, gz, gz, 0);
# endif
    __builtin_amdgcn_s_wait_tensorcnt(0);
#else
    for (int i = tid; i < NUM_G * 4; i += 32) sGT[i] = gptr[i];
#endif
  }
}

// ---------------------------------------------------------------------------
__global__ void __launch_bounds__(256)
init_kernel(unsigned int* w, int n) {
  int i = blockIdx.x * blockDim.x + threadIdx.x;
  int stride = gridDim.x * blockDim.x;
  for (; i < n; i += stride) w[i] = 0u;
}

// ---------------------------------------------------------------------------
// Pass 1: per-anchor best-GT (iou, idx) and per-GT best-anchor argmax.
// grid = (A/1024, B), 256 threads, 4 anchors/thread.
__global__ void __launch_bounds__(256)
match_kernel(const float* __restrict__ bbox, const float* __restrict__ gt,
             unsigned long long* __restrict__ bestPack,
             unsigned long long* __restrict__ perGT) {
  __shared__ float sGT[NUM_G * 4];
  __shared__ unsigned long long sBest[NUM_G];

  const int b   = blockIdx.y;
  const int tid = threadIdx.x;
  stage_gt_tile(gt + (size_t)b * NUM_G * 4, sGT, tid);
  if (tid < NUM_G) sBest[tid] = 0ull;
  __syncthreads();

  const int aBase = blockIdx.x * 1024;
  const float4* bb4 = (const float4*)(bbox + (size_t)b * NUM_A * 4);

  float4 pb[4];
  float  areaA[4];
#pragma unroll
  for (int i = 0; i < 4; ++i) {
    int a = aBase + tid + i * 256;
    pb[i] = bb4[a];
    areaA[i] = (pb[i].z - pb[i].x) * (pb[i].w - pb[i].y);
  }

  float bestIoU[4] = {-1.f, -1.f, -1.f, -1.f};
  int   bestG[4]   = {0, 0, 0, 0};
  const int lane = tid & 31;

  for (int g = 0; g < NUM_G; ++g) {
    float gx1 = sGT[g * 4 + 0], gy1 = sGT[g * 4 + 1];
    float gx2 = sGT[g * 4 + 2], gy2 = sGT[g * 4 + 3];
    float areaG = (gx2 - gx1) * (gy2 - gy1);
    float    wIoU = -1.f;      // this thread's best anchor for GT g
    unsigned wA   = 0u;
#pragma unroll
    for (int i = 0; i < 4; ++i) {
      float ix1 = fmaxf(pb[i].x, gx1), iy1 = fmaxf(pb[i].y, gy1);
      float ix2 = fminf(pb[i].z, gx2), iy2 = fminf(pb[i].w, gy2);
      float iw = fmaxf(ix2 - ix1, 0.f), ih = fmaxf(iy2 - iy1, 0.f);
      float inter = iw * ih;
      float iou = inter * fast_rcp(areaA[i] + areaG - inter + EPSF);
      if (iou > bestIoU[i]) { bestIoU[i] = iou; bestG[i] = g; }
      if (iou > wIoU) { wIoU = iou; wA = (unsigned)(aBase + tid + i * 256); }
    }
    // cross-lane f32 max, then pick the lowest tying lane's anchor
    float m = wIoU;
#pragma unroll
    for (int off = 16; off > 0; off >>= 1) m = fmaxf(m, __shfl_xor(m, off));
    unsigned long long ball = __ballot(wIoU == m);
    int src = __ffsll(ball) - 1;
    unsigned aWin = __shfl(wA, src);
    if (lane == 0) {
      unsigned long long pk =
          ((unsigned long long)__float_as_uint(fmaxf(m, 0.f)) << 32) |
          (0xFFFFFFFFu - aWin);
      atomicMax(&sBest[g], pk);                   // ds_max_u64
    }
  }
  __syncthreads();
  if (tid < NUM_G) atomicMax(&perGT[b * NUM_G + tid], sBest[tid]);

#pragma unroll
  for (int i = 0; i < 4; ++i) {
    int a = aBase + tid + i * 256;
    bestPack[(size_t)b * NUM_A + a] =
        ((unsigned long long)__float_as_uint(bestIoU[i]) << 32) |
        (unsigned)bestG[i];
  }
}

// ---------------------------------------------------------------------------
// Pass 2a: mark forced positives (best anchor per GT).
__global__ void __launch_bounds__(256)
force_kernel(const unsigned long long* __restrict__ perGT,
             unsigned char* __restrict__ posForce) {
  int i = blockIdx.x * blockDim.x + threadIdx.x;
  if (i < NUM_B * NUM_G) {
    int b = i >> 6;
    unsigned a = 0xFFFFFFFFu - (unsigned)(perGT[i] & 0xFFFFFFFFull);
    if (a < NUM_A) posForce[(size_t)b * NUM_A + a] = 1;
  }
}

// ---------------------------------------------------------------------------
// Pass 2b: focal + DIoU + per-image histogram. grid = (16, B), 4096 anch/block.
__global__ void __launch_bounds__(256)
loss_kernel(const float* __restrict__ bbox, const float* __restrict__ conf,
            const float* __restrict__ gt,
            const unsigned long long* __restrict__ bestPack,
            const unsigned char* __restrict__ posForce,
            float* __restrict__ numPos, float* __restrict__ locSum,
            float* __restrict__ posFocal,
            unsigned int* __restrict__ histCnt, float* __restrict__ histSum) {
  __shared__ unsigned int hC[NBUCKET];
  __shared__ float        hS[NBUCKET];
  __shared__ float        sGT[NUM_G * 4];
  const int b = blockIdx.y, tid = threadIdx.x;
  stage_gt_tile(gt + (size_t)b * NUM_G * 4, sGT, tid);
  for (int i = tid; i < NBUCKET; i += 256) { hC[i] = 0u; hS[i] = 0.f; }
  __syncthreads();

  float tLoc = 0.f, tPos = 0.f, tFl = 0.f;
  const int aBase = blockIdx.x * 4096;

  for (int it = 0; it < 16; ++it) {
    int a = aBase + tid + it * 256;
    size_t idx = (size_t)b * NUM_A + a;
    unsigned long long pk = bestPack[idx];
    float iou = __uint_as_float((unsigned)(pk >> 32));
    int g = (int)(pk & 0xFFFFFFFFull);
    bool pos = (iou > 0.5f) || (posForce[idx] != 0);
    float t = pos ? 1.f : 0.f;
    float l = conf[idx];

    // focal (stable BCE-with-logits) using v_exp_f32 / v_log_f32
    float ce = fmaxf(l, 0.f) - l * t + __logf(1.f + __expf(-fabsf(l)));
    float p  = fast_rcp(1.f + __expf(-l));
    float pt = p * t + (1.f - p) * (1.f - t);
    pt = fminf(fmaxf(pt, EPSF), 1.f - EPSF);
    float w  = (1.f - pt) * (1.f - pt);
    float at = 0.25f * t + 0.75f * (1.f - t);
    float fl = at * w * ce;

    if (pos) {
      float4 pbx = ((const float4*)bbox)[idx];
      float gx1 = sGT[g * 4 + 0], gy1 = sGT[g * 4 + 1];
      float gx2 = sGT[g * 4 + 2], gy2 = sGT[g * 4 + 3];
      float ap = (pbx.z - pbx.x) * (pbx.w - pbx.y);
      float ag = (gx2 - gx1) * (gy2 - gy1);
      float ix1 = fmaxf(pbx.x, gx1), iy1 = fmaxf(pbx.y, gy1);
      float ix2 = fminf(pbx.z, gx2), iy2 = fminf(pbx.w, gy2);
      float iw = fmaxf(ix2 - ix1, 0.f), ih = fmaxf(iy2 - iy1, 0.f);
      float inter = iw * ih;
      float iou2 = inter * fast_rcp(ap + ag - inter + EPSF);
      float dx = (pbx.x + pbx.z) * 0.5f - (gx1 + gx2) * 0.5f;
      float dy = (pbx.y + pbx.w) * 0.5f - (gy1 + gy2) * 0.5f;
      float cd2 = dx * dx + dy * dy;
      float ddx = fmaxf(pbx.z, gx2) - fminf(pbx.x, gx1);
      float ddy = fmaxf(pbx.w, gy2) - fminf(pbx.y, gy1);
      float diag2 = ddx * ddx + ddy * ddy + EPSF;
      tLoc += 1.f - iou2 + cd2 * fast_rcp(diag2);
      tFl  += fl;
      tPos += 1.f;
    } else {
      unsigned bkt = __float_as_uint(fl) >> 19;   // fl >= 0 -> 12-bit key
      atomicAdd(&hC[bkt], 1u);
      atomicAdd(&hS[bkt], fl);
    }
  }

  // wave32 reductions via WMMA (EXEC all-ones: uniform point, full waves)
  float sLoc = wave_sum32(tLoc);
  float sFl  = wave_sum32(tFl);
  float sPos = wave_sum32(tPos);
  if ((tid & 31) == 0) {
    atomicAdd(&locSum[b], sLoc);
    atomicAdd(&posFocal[b], sFl);
    atomicAdd(&numPos[b], sPos);
  }
  __syncthreads();
  for (int i = tid; i < NBUCKET; i += 256) {
    unsigned c = hC[i];
    if (c) {
      atomicAdd(&histCnt[b * NBUCKET + i], c);
      atomicAdd(&histSum[b * NBUCKET + i], hS[i]);
    }
  }
}

// ---------------------------------------------------------------------------
// Pass 3: per-image top-k negative sum via histogram suffix scan. grid = B.
__global__ void __launch_bounds__(256)
select_kernel(const unsigned int* __restrict__ histCnt,
              const float* __restrict__ histSum,
              const float* __restrict__ numPos,
              const float* __restrict__ posFocal,
              float* __restrict__ confPer) {
  __shared__ int   sCnt[256];
  __shared__ float sSum[256];
  const int b = blockIdx.x, tid = threadIdx.x;
  float npF = numPos[b];
  int np = (int)(npF + 0.5f);
  int k = min(NUM_A - np, np * 3);

  int lc[16]; float ls[16];
  int cCnt = 0; float cSum = 0.f;
#pragma unroll
  for (int j = 0; j < 16; ++j) {
    int bkt = (NBUCKET - 1) - (tid * 16 + j);     // descending value order
    int   c = (int)histCnt[b * NBUCKET + bkt];
    float s = histSum[b * NBUCKET + bkt];
    lc[j] = c; ls[j] = s; cCnt += c; cSum += s;
  }
  sCnt[tid] = cCnt; sSum[tid] = cSum;
  __syncthreads();
  for (int off = 1; off < 256; off <<= 1) {       // inclusive scan
    int c = sCnt[tid]; float s = sSum[tid];
    int cl = 0; float sl = 0.f;
    if (tid >= off) { cl = sCnt[tid - off]; sl = sSum[tid - off]; }
    __syncthreads();
    sCnt[tid] = c + cl; sSum[tid] = s + sl;
    __syncthreads();
  }
  int cumC = sCnt[tid];
  int befC = cumC - cCnt;
  float befS = sSum[tid] - cSum;

  if (k <= 0) {
    if (tid == 0) confPer[b] = posFocal[b] / fmaxf(npF, 1.0f);
  } else if (befC < k && k <= cumC) {
    int running = befC; float acc = befS; float negSum = 0.f;
#pragma unroll
    for (int j = 0; j < 16; ++j) {
      int c = lc[j];
      if (running + c >= k) {
        int rem = k - running;
        float avg = (c > 0) ? ls[j] / (float)c : 0.f;
        negSum = acc + (float)rem * avg;          // mean-corrected boundary
        break;
      }
      running += c; acc += ls[j];
    }
    confPer[b] = (posFocal[b] + negSum) / fmaxf(npF + (float)k, 1.f);
  }
}

// ---------------------------------------------------------------------------
__global__ void __launch_bounds__(32)
final_kernel(const float* __restrict__ numPos, const float* __restrict__ locSum,
             const float* __restrict__ confPer, float* __restrict__ out) {
  int tid = threadIdx.x;
  float np = 0.f, lc = 0.f, cf = 0.f;
  if (tid < NUM_B) { np = numPos[tid]; lc = locSum[tid]; cf = confPer[tid]; }
  for (int off = 16; off > 0; off >>= 1) {
    np += __shfl_xor(np, off);
    lc += __shfl_xor(lc, off);
    cf += __shfl_xor(cf, off);
  }
  if (tid == 0) {
    float tp = fmaxf(np, 1.f);
    float avgLoc = lc / tp, avgConf = cf / tp;
    out[0] = avgLoc + avgConf;
    out[1] = avgConf;
    out[2] = avgLoc;
  }
}

// ---------------------------------------------------------------------------
extern "C" void kernel_launch(void* const* d_in, const int* in_sizes, int n_in,
                              void* d_out, int out_size, void* d_ws,
                              size_t ws_size, hipStream_t stream) {
  (void)in_sizes; (void)n_in; (void)out_size; (void)ws_size;
  const float* bbox = (const float*)d_in[0];   // [16,65536,4]
  const float* conf = (const float*)d_in[1];   // [16,65536]
  const float* gt   = (const float*)d_in[2];   // [16,64,4]
  float* out = (float*)d_out;                  // [3]
  char* ws = (char*)d_ws;

  unsigned long long* perGT   = (unsigned long long*)(ws + PERGT_OFF);
  float*              numPos  = (float*)(ws + NUMPOS_OFF);
  float*              locSum  = (float*)(ws + LOCSUM_OFF);
  float*              posFoc  = (float*)(ws + POSFOC_OFF);
  float*              confPer = (float*)(ws + CONF_OFF);
  unsigned int*       histCnt = (unsigned int*)(ws + HISTC_OFF);
  float*              histSum = (float*)(ws + HISTS_OFF);
  unsigned char*      posForce= (unsigned char*)(ws + PFORCE_OFF);
  unsigned long long* bestPack= (unsigned long long*)(ws + BESTP_OFF);

  init_kernel<<<(ZERO_WORDS + 255) / 256, 256, 0, stream>>>(
      (unsigned int*)ws, (int)ZERO_WORDS);
  match_kernel<<<dim3(NUM_A / 1024, NUM_B), 256, 0, stream>>>(
      bbox, gt, bestPack, perGT);
  force_kernel<<<(NUM_B * NUM_G + 255) / 256, 256, 0, stream>>>(
      perGT, posForce);
  loss_kernel<<<dim3(NUM_A / 4096, NUM_B), 256, 0, stream>>>(
      bbox, conf, gt, bestPack, posForce, numPos, locSum, posFoc,
      histCnt, histSum);
  select_kernel<<<NUM_B, 256, 0, stream>>>(
      histCnt, histSum, numPos, posFoc, confPer);
  final_kernel<<<1, 32, 0, stream>>>(numPos, locSum, confPer, out);
}